// SGC_74045236183293
// MI455X (gfx1250) — compile-verified
//
#include <hip/hip_runtime.h>

// ---------------------------------------------------------------------------
// SGC on gfx1250 (MI455X):
//   deg -> rsqrt -> propagate (x->h1) -> propagate (h1->h2) -> WMMA GEMM + bias
// Propagation uses hardware global_atomic_add_f32 (unsafeAtomicAdd).
// GEMM uses V_WMMA_F32_16X16X4_F32 (f32 A/B, f32 accum) -> exact f32 math.
// ---------------------------------------------------------------------------

typedef __attribute__((ext_vector_type(2))) float v2f;
typedef __attribute__((ext_vector_type(8))) float v8f;

#define IN_CH 48
#define OUT_CH 64

// ---------------- zero scratch ----------------
__global__ void sgc_zero_f32(float* __restrict__ p, int n) {
    int i = blockIdx.x * blockDim.x + threadIdx.x;
    if (i < n) p[i] = 0.0f;
}

// ---------------- degree over destination (col) indices ----------------
__global__ void sgc_degree(const int* __restrict__ col, float* __restrict__ deg, int E) {
    int e = blockIdx.x * blockDim.x + threadIdx.x;
    if (e < E) {
        unsafeAtomicAdd(&deg[col[e]], 1.0f);   // global_atomic_add_f32
    }
}

// ---------------- deg -> deg^{-1/2} in place (0 if deg==0) ----------------
__global__ void sgc_rsqrt(float* __restrict__ deg, int n) {
    int i = blockIdx.x * blockDim.x + threadIdx.x;
    if (i < n) {
        float d = deg[i];
        deg[i] = (d > 0.0f) ? rsqrtf(d) : 0.0f;
    }
}

// ---------------- one propagation round: dst[col] += src[row] * norm ------
__global__ void sgc_propagate(const float* __restrict__ src,
                              float* __restrict__ dst,
                              const int* __restrict__ row,
                              const int* __restrict__ col,
                              const float* __restrict__ dis,
                              int E) {
    int e = blockIdx.x * blockDim.x + threadIdx.x;
    if (e >= E) return;
    int r = row[e];
    int c = col[e];
    float norm = dis[r] * dis[c];
    const float4* s4 = reinterpret_cast<const float4*>(src + (size_t)r * IN_CH);
    float* d = dst + (size_t)c * IN_CH;
#pragma unroll
    for (int j = 0; j < IN_CH / 4; ++j) {
        float4 v = s4[j];
        unsafeAtomicAdd(d + 4 * j + 0, v.x * norm);
        unsafeAtomicAdd(d + 4 * j + 1, v.y * norm);
        unsafeAtomicAdd(d + 4 * j + 2, v.z * norm);
        unsafeAtomicAdd(d + 4 * j + 3, v.w * norm);
    }
}

// ---------------- out = H[M,48] @ W[48,64] + b, via V_WMMA_F32_16X16X4_F32 --
// One wave per 16-row tile; 4 accumulators cover N=64.
// Layouts per CDNA5 ISA 7.12.2:
//   A 16x4 f32  : lane%16 = M row; VGPR{0,1} = K {kh, kh+1}, kh = 2*(lane/16)
//   B 4x16 f32  : lane%16 = N col; VGPR{0,1} = K {kh, kh+1}, kh = 2*(lane/16)
//   C/D 16x16   : VGPR v -> M = v + 8*(lane/16); N = lane%16
__global__ void sgc_gemm_wmma(const float* __restrict__ H,
                              const float* __restrict__ Wg,
                              const float* __restrict__ Bg,
                              float* __restrict__ out,
                              int mtiles) {
    __shared__ float Ws[IN_CH * OUT_CH];   // 12 KB
    __shared__ float Bs[OUT_CH];

    for (int i = threadIdx.x; i < IN_CH * OUT_CH; i += blockDim.x) Ws[i] = Wg[i];
    if (threadIdx.x < OUT_CH) Bs[threadIdx.x] = Bg[threadIdx.x];
    __syncthreads();

    const int wave = threadIdx.x >> 5;
    const int lane = threadIdx.x & 31;
    const int tile = blockIdx.x * (blockDim.x >> 5) + wave;
    if (tile >= mtiles) return;            // wave-uniform: EXEC stays all-ones

    const int mbase = tile * 16;
    const int lh = lane & 15;              // row of A / col of B within tile
    const int kh = (lane >> 4) * 2;        // K sub-offset selected by lane half
    const int m  = mbase + lh;

    v8f acc0 = {}, acc1 = {}, acc2 = {}, acc3 = {};

#pragma unroll
    for (int k0 = 0; k0 < IN_CH; k0 += 4) {
        v2f a;
        a.x = H[(size_t)m * IN_CH + k0 + kh + 0];
        a.y = H[(size_t)m * IN_CH + k0 + kh + 1];

        v2f b0, b1, b2, b3;
        b0.x = Ws[(k0 + kh + 0) * OUT_CH +  0 + lh];
        b0.y = Ws[(k0 + kh + 1) * OUT_CH +  0 + lh];
        b1.x = Ws[(k0 + kh + 0) * OUT_CH + 16 + lh];
        b1.y = Ws[(k0 + kh + 1) * OUT_CH + 16 + lh];
        b2.x = Ws[(k0 + kh + 0) * OUT_CH + 32 + lh];
        b2.y = Ws[(k0 + kh + 1) * OUT_CH + 32 + lh];
        b3.x = Ws[(k0 + kh + 0) * OUT_CH + 48 + lh];
        b3.y = Ws[(k0 + kh + 1) * OUT_CH + 48 + lh];

        acc0 = __builtin_amdgcn_wmma_f32_16x16x4_f32(false, a, false, b0, (short)0, acc0, false, false);
        acc1 = __builtin_amdgcn_wmma_f32_16x16x4_f32(false, a, false, b1, (short)0, acc1, false, false);
        acc2 = __builtin_amdgcn_wmma_f32_16x16x4_f32(false, a, false, b2, (short)0, acc2, false, false);
        acc3 = __builtin_amdgcn_wmma_f32_16x16x4_f32(false, a, false, b3, (short)0, acc3, false, false);
    }

    const int mh = (lane >> 4) * 8;        // C/D row half select
#pragma unroll
    for (int v = 0; v < 8; ++v) {
        const int rowi = mbase + v + mh;
        float* o = out + (size_t)rowi * OUT_CH;
        o[ 0 + lh] = acc0[v] + Bs[ 0 + lh];
        o[16 + lh] = acc1[v] + Bs[16 + lh];
        o[32 + lh] = acc2[v] + Bs[32 + lh];
        o[48 + lh] = acc3[v] + Bs[48 + lh];
    }
}

// ---------------------------------------------------------------------------
extern "C" void kernel_launch(void* const* d_in, const int* in_sizes, int n_in,
                              void* d_out, int out_size, void* d_ws, size_t ws_size,
                              hipStream_t stream) {
    const float* x  = (const float*)d_in[0];
    const int*   ei = (const int*)d_in[1];
    const float* W  = (const float*)d_in[2];
    const float* b  = (const float*)d_in[3];
    float* out = (float*)d_out;

    const int M = in_sizes[0] / IN_CH;     // 100000 nodes
    const int E = in_sizes[1] / 2;         // 1.6M edges
    const int* row = ei;                   // edge_index[0, :]
    const int* col = ei + E;               // edge_index[1, :]

    // workspace: [deg/dis : M][h1 : M*48][h2 : M*48]
    float* ws  = (float*)d_ws;
    float* deg = ws;
    float* h1  = ws + M;
    float* h2  = h1 + (size_t)M * IN_CH;
    const int nzero = M + 2 * M * IN_CH;

    const int T = 256;
    sgc_zero_f32<<<(nzero + T - 1) / T, T, 0, stream>>>(ws, nzero);
    sgc_degree<<<(E + T - 1) / T, T, 0, stream>>>(col, deg, E);
    sgc_rsqrt<<<(M + T - 1) / T, T, 0, stream>>>(deg, M);
    sgc_propagate<<<(E + T - 1) / T, T, 0, stream>>>(x,  h1, row, col, deg, E);
    sgc_propagate<<<(E + T - 1) / T, T, 0, stream>>>(h1, h2, row, col, deg, E);

    const int mtiles = M / 16;             // 6250 (M is a multiple of 16)
    const int wavesPerBlock = T / 32;      // 8 tiles per block
    sgc_gemm_wmma<<<(mtiles + wavesPerBlock - 1) / wavesPerBlock, T, 0, stream>>>(
        h2, W, b, out, mtiles);
}